// PointContrastive_10892037063168
// MI455X (gfx1250) — compile-verified
//
#include <hip/hip_runtime.h>
#include <hip/hip_bf16.h>
#include <math.h>

typedef __attribute__((ext_vector_type(2))) float v2f;
typedef __attribute__((ext_vector_type(8))) float v8f;

#define BS 16
#define P  4096
#define M  32
#define D  768
#define N  512   /* BS*M */

// ---------------------------------------------------------------------------
// Block reduction helpers (256 threads)
// ---------------------------------------------------------------------------
__device__ __forceinline__ float block_reduce_sum(float v, float* red) {
    red[threadIdx.x] = v;
    __syncthreads();
    for (int off = 128; off > 0; off >>= 1) {
        if ((int)threadIdx.x < off) red[threadIdx.x] += red[threadIdx.x + off];
        __syncthreads();
    }
    float r = red[0];
    __syncthreads();
    return r;
}

__device__ __forceinline__ float block_reduce_max(float v, float* red) {
    red[threadIdx.x] = v;
    __syncthreads();
    for (int off = 128; off > 0; off >>= 1) {
        if ((int)threadIdx.x < off)
            red[threadIdx.x] = fmaxf(red[threadIdx.x], red[threadIdx.x + off]);
        __syncthreads();
    }
    float r = red[0];
    __syncthreads();
    return r;
}

// ---------------------------------------------------------------------------
// Kernel 1: per-mask point counts  npts[n] = sum_p mask_pts[n, p]
// ---------------------------------------------------------------------------
__global__ void npts_kernel(const float* __restrict__ mask_pts,
                            float* __restrict__ npts) {
    __shared__ float red[256];
    const int row = blockIdx.x;                       // 0..511
    const float* src = mask_pts + (size_t)row * P;
    float s = 0.f;
    for (int i = threadIdx.x; i < P; i += 256) s += src[i];
    s = block_reduce_sum(s, red);
    if (threadIdx.x == 0) npts[row] = s;
}

// ---------------------------------------------------------------------------
// Kernel 2: segmented masked mean via f32 WMMA.
//   Per batch b: C[32, 768] = mask_pts[b][32, 4096] x pts[b][4096, 768]
// One wave owns one (b, 16-wide d-tile) and accumulates BOTH 16-mask tiles,
// so each net_out fragment is loaded once and feeds two v_wmma ops.
// Epilogue multiplies by rcp(npts + 1e-12) and writes avg_feats.
// ---------------------------------------------------------------------------
__global__ void seg_gemm_kernel(const float* __restrict__ net_out,
                                const float* __restrict__ mask_pts,
                                const float* __restrict__ npts,
                                float* __restrict__ avg) {
    const int wave = (int)((blockIdx.x * blockDim.x + threadIdx.x) >> 5); // 0..767
    const int lane = (int)(threadIdx.x & 31);
    const int b    = wave / (D / 16);   // 0..15
    const int dT   = wave % (D / 16);   // 0..47
    const int d0   = dT * 16;
    const int hl   = lane >> 4;         // 0 or 1 (K pair select)
    const int li   = lane & 15;         // M (for A) / N (for B)

    // A: mask rows, two contiguous K values per lane -> one 8B load
    const float* mrow0 = mask_pts + ((size_t)(b * M + li)) * P + 2 * hl;  // m = li
    const float* mrow1 = mrow0 + (size_t)16 * P;                          // m = 16+li
    // B: pts[p, d0+li], rows k = 2*hl, 2*hl+1  (stride D between K values)
    const float* bcol  = net_out + ((size_t)b * P + 2 * hl) * D + d0 + li;

    v8f c0 = {};
    v8f c1 = {};
    #pragma unroll 4
    for (int p0 = 0; p0 < P; p0 += 4) {
        if ((p0 & 255) == 0) {
            // near-scope (WGP) prefetch of the streamed mask rows
            __builtin_prefetch((const void*)(mrow0 + p0 + 512), 0, 3);
            __builtin_prefetch((const void*)(mrow1 + p0 + 512), 0, 3);
        }
        v2f a0 = *(const v2f*)(mrow0 + p0);
        v2f a1 = *(const v2f*)(mrow1 + p0);
        v2f bb;
        bb[0] = bcol[(size_t)(p0 + 0) * D];
        bb[1] = bcol[(size_t)(p0 + 1) * D];
        c0 = __builtin_amdgcn_wmma_f32_16x16x4_f32(false, a0, false, bb,
                                                   (short)0, c0, false, false);
        c1 = __builtin_amdgcn_wmma_f32_16x16x4_f32(false, a1, false, bb,
                                                   (short)0, c1, false, false);
    }

    // C layout: VGPR r holds (M = r + 8*hl, N = li)
    #pragma unroll
    for (int r = 0; r < 8; ++r) {
        const int m0 = r + 8 * hl;
        const int g0 = b * M + m0;          // masks 0..15
        const int g1 = g0 + 16;             // masks 16..31
        const float rcp0 = __builtin_amdgcn_rcpf(npts[g0] + 1e-12f);
        const float rcp1 = __builtin_amdgcn_rcpf(npts[g1] + 1e-12f);
        avg[(size_t)g0 * D + d0 + li] = c0[r] * rcp0;
        avg[(size_t)g1 * D + d0 + li] = c1[r] * rcp1;
    }
}

// ---------------------------------------------------------------------------
// Kernel 3: logits[512,512] = (mask_embs @ avg^T) * exp(logit_scale)
// One wave per 16x16 output tile, K = 768 in steps of 4.
// Both A and B fragments are contiguous-K -> 8B vector loads.
// ---------------------------------------------------------------------------
__global__ void logits_kernel(const float* __restrict__ mask_embs,
                              const float* __restrict__ avg,
                              const float* __restrict__ logit_scale,
                              float* __restrict__ logits) {
    const int wave = (int)((blockIdx.x * blockDim.x + threadIdx.x) >> 5); // 0..1023
    const int lane = (int)(threadIdx.x & 31);
    const int rT = wave >> 5;     // 0..31 row tile
    const int cT = wave & 31;     // 0..31 col tile
    const int hl = lane >> 4;
    const int li = lane & 15;

    const float* arow = mask_embs + ((size_t)(rT * 16 + li)) * D + 2 * hl;
    const float* brow = avg       + ((size_t)(cT * 16 + li)) * D + 2 * hl;

    v8f c = {};
    #pragma unroll 8
    for (int k0 = 0; k0 < D; k0 += 4) {
        v2f a  = *(const v2f*)(arow + k0);
        v2f bb = *(const v2f*)(brow + k0);
        c = __builtin_amdgcn_wmma_f32_16x16x4_f32(false, a, false, bb,
                                                  (short)0, c, false, false);
    }

    const float scale = expf(logit_scale[0]);
    #pragma unroll
    for (int r = 0; r < 8; ++r) {
        const int i = rT * 16 + r + 8 * hl;
        const int j = cT * 16 + li;
        logits[(size_t)i * N + j] = c[r] * scale;
    }
}

// ---------------------------------------------------------------------------
// Kernel 4: row (blocks 0..511) and column (blocks 512..1023) log-sum-exp.
//   losses[i]     = lse_row(i) - logits[i,i]   (texts_loss, axis=1)
//   losses[N + j] = lse_col(j) - logits[j,j]   (pts_loss,   axis=0)
// Zeroed when npts[idx] == 0 (target -100 / ignored).
// ---------------------------------------------------------------------------
__global__ void lse_kernel(const float* __restrict__ logits,
                           const float* __restrict__ npts,
                           float* __restrict__ losses) {
    __shared__ float red[256];
    const int id = blockIdx.x;              // 0..1023
    const bool isRow = id < N;
    const int idx = isRow ? id : id - N;

    float mx = -INFINITY;
    for (int j = threadIdx.x; j < N; j += 256) {
        float v = isRow ? logits[(size_t)idx * N + j] : logits[(size_t)j * N + idx];
        mx = fmaxf(mx, v);
    }
    mx = block_reduce_max(mx, red);

    float s = 0.f;
    for (int j = threadIdx.x; j < N; j += 256) {
        float v = isRow ? logits[(size_t)idx * N + j] : logits[(size_t)j * N + idx];
        s += expf(v - mx);
    }
    s = block_reduce_sum(s, red);

    if (threadIdx.x == 0) {
        const float lse  = mx + logf(s);
        const float diag = logits[(size_t)idx * N + idx];
        losses[id] = (npts[idx] > 0.f) ? (lse - diag) : 0.f;
    }
}

// ---------------------------------------------------------------------------
// Kernel 5: final scalar = (nonzero_mean(texts) + nonzero_mean(pts)) / 2
// ---------------------------------------------------------------------------
__global__ void final_kernel(const float* __restrict__ losses,
                             float* __restrict__ out) {
    __shared__ float red[256];
    float st = 0.f, ct = 0.f, sp = 0.f, cp = 0.f;
    for (int i = threadIdx.x; i < N; i += 256) {
        float t = losses[i];
        float p = losses[N + i];
        if (t > 0.f) { st += t; ct += 1.f; }
        if (p > 0.f) { sp += p; cp += 1.f; }
    }
    st = block_reduce_sum(st, red);
    ct = block_reduce_sum(ct, red);
    sp = block_reduce_sum(sp, red);
    cp = block_reduce_sum(cp, red);
    if (threadIdx.x == 0) {
        const float mt = (ct > 0.f) ? st / fmaxf(ct, 1.f) : 0.f;
        const float mp = (cp > 0.f) ? sp / fmaxf(cp, 1.f) : 0.f;
        out[0] = 0.5f * (mt + mp);
    }
}

// ---------------------------------------------------------------------------
extern "C" void kernel_launch(void* const* d_in, const int* in_sizes, int n_in,
                              void* d_out, int out_size, void* d_ws, size_t ws_size,
                              hipStream_t stream) {
    const float* net_out     = (const float*)d_in[0];
    /* d_in[1] = pt_offset (regular layout; unused) */
    const float* mask_embs   = (const float*)d_in[2];
    const float* mask_pts    = (const float*)d_in[3];
    const float* logit_scale = (const float*)d_in[4];
    float* out = (float*)d_out;

    float* ws     = (float*)d_ws;
    float* npts   = ws;                                 // 512
    float* avg    = npts + N;                           // 512*768
    float* logits = avg + (size_t)N * D;                // 512*512
    float* losses = logits + (size_t)N * N;             // 1024

    npts_kernel  <<<N, 256, 0, stream>>>(mask_pts, npts);
    seg_gemm_kernel<<<(BS * (D / 16)) / 8, 256, 0, stream>>>(net_out, mask_pts,
                                                             npts, avg);
    logits_kernel<<<((N / 16) * (N / 16)) / 8, 256, 0, stream>>>(mask_embs, avg,
                                                                 logit_scale, logits);
    lse_kernel   <<<2 * N, 256, 0, stream>>>(logits, npts, losses);
    final_kernel <<<1, 256, 0, stream>>>(losses, out);
}